// Pseudo3DConv_25383256719968
// MI455X (gfx1250) — compile-verified
//
#include <hip/hip_runtime.h>
#include <hip/hip_bf16.h>
#include <math.h>

#define NPTS 500
#define NP12 12
#define NPP  4

typedef float v2f __attribute__((ext_vector_type(2)));
typedef float v8f __attribute__((ext_vector_type(8)));

#define WMMA_F32(A, B, C) \
  __builtin_amdgcn_wmma_f32_16x16x4_f32(false, (A), false, (B), (short)0, (C), false, false)

// ---------------- prep: pad W_pconv1 (64x3 -> 64x4) and transpose clouds to (4,N), row3 = 0 ----------------
__global__ void prep_kernel(const float* __restrict__ W1,
                            const float* __restrict__ cloud,
                            const float* __restrict__ cloud_tar,
                            float* __restrict__ Wp,      // 64x4
                            float* __restrict__ c_t,     // 4x500
                            float* __restrict__ ct_t) {  // 4x500
  int t = blockIdx.x * blockDim.x + threadIdx.x;
  if (t < 64 * 4) {
    int m = t >> 2, k = t & 3;
    Wp[t] = (k < 3) ? W1[m * 3 + k] : 0.f;
    return;
  }
  t -= 64 * 4;
  if (t < 4 * NPTS) {
    int d = t / NPTS, n = t % NPTS;
    c_t[t]  = (d < 3) ? cloud[n * 3 + d] : 0.f;
    ct_t[t] = (d < 3) ? cloud_tar[n * 3 + d] : 0.f;
  }
}

// ---------------- brute-force KNN, 3 tasks, tie order matches top_k ----------------
__global__ void knn_kernel(const float* __restrict__ cloud,      // c  : (500,3)
                           const float* __restrict__ cloud_tar,  // ct : (500,3)
                           int* __restrict__ inds,               // ct->c, k=12
                           int* __restrict__ inds_self,          // ct->ct, k=12
                           int* __restrict__ inds_pp) {          // c->ct, k=4
  int task = blockIdx.y;
  int q = blockIdx.x * blockDim.x + threadIdx.x;
  if (q >= NPTS) return;
  const float* Q; const float* R; int k; int* out;
  if (task == 0)      { Q = cloud_tar; R = cloud;     k = NP12; out = inds; }
  else if (task == 1) { Q = cloud_tar; R = cloud_tar; k = NP12; out = inds_self; }
  else                { Q = cloud;     R = cloud_tar; k = NPP;  out = inds_pp; }
  float qx = Q[q*3], qy = Q[q*3+1], qz = Q[q*3+2];
  float bd[NP12]; int bi[NP12];
  for (int i = 0; i < NP12; ++i) { bd[i] = 3.4028235e38f; bi[i] = 0; }
  for (int r = 0; r < NPTS; ++r) {
    float dx = qx - R[r*3], dy = qy - R[r*3+1], dz = qz - R[r*3+2];
    float d2 = dx*dx + dy*dy + dz*dz;
    if (d2 < bd[k-1]) {                 // strict: equal keeps earlier index (top_k stable)
      int pos = k - 1;
      while (pos > 0 && bd[pos-1] > d2) { bd[pos] = bd[pos-1]; bi[pos] = bi[pos-1]; --pos; }
      bd[pos] = d2; bi[pos] = r;
    }
  }
  for (int j = 0; j < k; ++j) out[q*k + j] = bi[j];
}

// ---------------- global softmax of -||ct[n] - R[idx]|| over N*12 elems ----------------
__global__ void __launch_bounds__(1024)
softmax_w_kernel(const float* __restrict__ cloud, const float* __restrict__ cloud_tar,
                 const int* __restrict__ inds, const int* __restrict__ inds_self,
                 float* __restrict__ w_ct, float* __restrict__ w_cc) {
  const int M = NPTS * NP12;
  int task = blockIdx.x;
  const int* idx = task ? inds_self : inds;
  const float* R = task ? cloud_tar : cloud;
  float* w = task ? w_cc : w_ct;
  __shared__ float red[1024];
  int tid = threadIdx.x;

  float lmax = -3.4028235e38f;
  for (int j = tid; j < M; j += 1024) {
    int n = j / NP12, r = idx[j];
    float dx = cloud_tar[n*3+0] - R[r*3+0];
    float dy = cloud_tar[n*3+1] - R[r*3+1];
    float dz = cloud_tar[n*3+2] - R[r*3+2];
    float s = -sqrtf(dx*dx + dy*dy + dz*dz + 1e-12f);
    lmax = fmaxf(lmax, s);
  }
  red[tid] = lmax; __syncthreads();
  for (int st = 512; st > 0; st >>= 1) { if (tid < st) red[tid] = fmaxf(red[tid], red[tid+st]); __syncthreads(); }
  float gmax = red[0]; __syncthreads();

  float lsum = 0.f;
  for (int j = tid; j < M; j += 1024) {
    int n = j / NP12, r = idx[j];
    float dx = cloud_tar[n*3+0] - R[r*3+0];
    float dy = cloud_tar[n*3+1] - R[r*3+1];
    float dz = cloud_tar[n*3+2] - R[r*3+2];
    float s = -sqrtf(dx*dx + dy*dy + dz*dz + 1e-12f);
    lsum += expf(s - gmax);
  }
  red[tid] = lsum; __syncthreads();
  for (int st = 512; st > 0; st >>= 1) { if (tid < st) red[tid] += red[tid+st]; __syncthreads(); }
  float gsum = red[0];

  for (int j = tid; j < M; j += 1024) {
    int n = j / NP12, r = idx[j];
    float dx = cloud_tar[n*3+0] - R[r*3+0];
    float dy = cloud_tar[n*3+1] - R[r*3+1];
    float dz = cloud_tar[n*3+2] - R[r*3+2];
    float s = -sqrtf(dx*dx + dy*dy + dz*dz + 1e-12f);
    w[j] = expf(s - gmax) / gsum;
  }
}

// ---------------- fp32 WMMA GEMM, depth-2 pipelined, compile-time K ----------------
// Y[rowOff..rowOff+M) = act(W(MxK) @ X(KxN) + b). K % 4 == 0, M % 16 == 0, N = NPTS.
// One wave computes a 16x64 output strip: one A fragment feeds 4 independent WMMA chains.
// Fragments for step s are loaded while step s-2 is multiplied (2 WMMA groups of slack).
template <int K, int ACT>
__global__ void __launch_bounds__(256)
gemm_wmma_kernel(const float* __restrict__ W, const float* __restrict__ X,
                 const float* __restrict__ bias, float* __restrict__ Y,
                 int M, int rowOff) {
  constexpr int N = NPTS;
  constexpr int S = K / 4;            // number of 16x16x4 k-steps
  int wave = blockIdx.x * (blockDim.x >> 5) + (threadIdx.x >> 5);
  int mT = wave >> 3;                 // 8 strips of 64 cols cover 512 padded cols
  int nS = wave & 7;
  if (mT * 16 >= M) return;

  int lane = threadIdx.x & 31;
  int half = lane >> 4;               // 0: lanes 0-15, 1: lanes 16-31
  int l16  = lane & 15;
  int m  = mT * 16 + l16;
  int n0 = nS * 64 + l16;
  int n1 = n0 + 16, n2 = n0 + 32, n3 = n0 + 48;
  int nc0 = (n0 < N) ? n0 : N - 1;    // clamp: garbage cols never stored
  int nc1 = (n1 < N) ? n1 : N - 1;
  int nc2 = (n2 < N) ? n2 : N - 1;
  int nc3 = (n3 < N) ? n3 : N - 1;

  // Per-lane pointers; B pair = immediate offsets {s*4N, s*4N+N} floats.
  const float* pw  = W + (size_t)m * K + 2 * half;   // A: VGPR0=K+0/K+2, VGPR1=+1
  const float* pb  = X + (2 * half) * N;
  const float* pb0 = pb + nc0;
  const float* pb1 = pb + nc1;
  const float* pb2 = pb + nc2;
  const float* pb3 = pb + nc3;

  v8f acc0 = {}, acc1 = {}, acc2 = {}, acc3 = {};

  // stage 0
  v2f a0v = *(const v2f*)pw;
  v2f b00 = {pb0[0], pb0[N]};
  v2f b01 = {pb1[0], pb1[N]};
  v2f b02 = {pb2[0], pb2[N]};
  v2f b03 = {pb3[0], pb3[N]};
  // stage 1
  v2f a1v, b10, b11, b12, b13;
  if constexpr (S > 1) {
    a1v = *(const v2f*)(pw + 4);
    b10 = {pb0[4 * N], pb0[5 * N]};
    b11 = {pb1[4 * N], pb1[5 * N]};
    b12 = {pb2[4 * N], pb2[5 * N]};
    b13 = {pb3[4 * N], pb3[5 * N]};
  }

#pragma unroll 2
  for (int s = 2; s < S; ++s) {
    pw += 4; pb0 += 4 * N; pb1 += 4 * N; pb2 += 4 * N; pb3 += 4 * N;
    // load stage s (pointers now at stage s-1; +4 / +4N immediates reach stage s)
    v2f a2v = *(const v2f*)(pw + 4);
    v2f b20 = {pb0[4 * N], pb0[5 * N]};
    v2f b21 = {pb1[4 * N], pb1[5 * N]};
    v2f b22 = {pb2[4 * N], pb2[5 * N]};
    v2f b23 = {pb3[4 * N], pb3[5 * N]};
    // multiply stage s-2
    acc0 = WMMA_F32(a0v, b00, acc0);
    acc1 = WMMA_F32(a0v, b01, acc1);
    acc2 = WMMA_F32(a0v, b02, acc2);
    acc3 = WMMA_F32(a0v, b03, acc3);
    a0v = a1v; b00 = b10; b01 = b11; b02 = b12; b03 = b13;
    a1v = a2v; b10 = b20; b11 = b21; b12 = b22; b13 = b23;
  }
  acc0 = WMMA_F32(a0v, b00, acc0);
  acc1 = WMMA_F32(a0v, b01, acc1);
  acc2 = WMMA_F32(a0v, b02, acc2);
  acc3 = WMMA_F32(a0v, b03, acc3);
  if constexpr (S > 1) {
    acc0 = WMMA_F32(a1v, b10, acc0);
    acc1 = WMMA_F32(a1v, b11, acc1);
    acc2 = WMMA_F32(a1v, b12, acc2);
    acc3 = WMMA_F32(a1v, b13, acc3);
  }

#pragma unroll
  for (int r = 0; r < 8; ++r) {       // C/D: VGPR r = rows r (lo half) and r+8 (hi half)
    int row = mT * 16 + r + 8 * half;
    float bv = bias[row];
    float v0 = acc0[r] + bv;
    float v1 = acc1[r] + bv;
    float v2 = acc2[r] + bv;
    float v3 = acc3[r] + bv;
    if constexpr (ACT) {
      v0 = (v0 > 0.f) ? v0 : 0.01f * v0;
      v1 = (v1 > 0.f) ? v1 : 0.01f * v1;
      v2 = (v2 > 0.f) ? v2 : 0.01f * v2;
      v3 = (v3 > 0.f) ? v3 : 0.01f * v3;
    }
    float* yrow = Y + (size_t)(rowOff + row) * N;
    if (n0 < N) yrow[n0] = v0;
    if (n1 < N) yrow[n1] = v1;
    if (n2 < N) yrow[n2] = v2;
    if (n3 < N) yrow[n3] = v3;
  }
}

// ---------------- stage1: sit/spt maxpool -> img_diff, cloud_diff ----------------
__global__ void gmax1_kernel(const float* __restrict__ A,    // ifeat(img_tar)  (128,500)
                             const float* __restrict__ B,    // ifeat(img)      (128,500)
                             const float* __restrict__ cf,   // pfeat(cloud_tar)
                             const float* __restrict__ ctf,  // pfeat(cloud)
                             const int* __restrict__ inds, const float* __restrict__ w_ct,
                             float* __restrict__ fc1_in,     // rows 0-127 <- img_diff
                             float* __restrict__ fc2_in) {   // rows 0-127 <- cloud_diff
  int t = blockIdx.x * blockDim.x + threadIdx.x;
  if (t >= 128 * NPTS) return;
  int ch = t / NPTS, n = t % NPTS;
  float sit = -3.4028235e38f, spt = -3.4028235e38f;
  for (int j = 0; j < NP12; ++j) {
    int ji = n * NP12 + j;
    int ib = inds[ji];
    float w = w_ct[ji];
    sit = fmaxf(sit, B[ch * NPTS + ib] * w);
    spt = fmaxf(spt, ctf[ch * NPTS + ib] * w);
  }
  fc1_in[ch * NPTS + n] = A[ch * NPTS + n] - sit;
  fc2_in[ch * NPTS + n] = cf[ch * NPTS + n] - spt;
}

// ---------------- stage2: spd/sid maxpool into the concat halves ----------------
__global__ void gmax2_kernel(const int* __restrict__ inds_self, const float* __restrict__ w_cc,
                             float* __restrict__ fc1_in,   // read rows 0-127 (img_diff), write 128-255 (spd)
                             float* __restrict__ fc2_in) { // read rows 0-127 (cloud_diff), write 128-255 (sid)
  int t = blockIdx.x * blockDim.x + threadIdx.x;
  if (t >= 128 * NPTS) return;
  int ch = t / NPTS, n = t % NPTS;
  float spd = -3.4028235e38f, sid = -3.4028235e38f;
  for (int j = 0; j < NP12; ++j) {
    int ji = n * NP12 + j;
    int ib = inds_self[ji];
    float w = w_cc[ji];
    spd = fmaxf(spd, fc2_in[ch * NPTS + ib] * w);  // cloud_diff
    sid = fmaxf(sid, fc1_in[ch * NPTS + ib] * w);  // img_diff
  }
  fc1_in[(128 + ch) * NPTS + n] = spd;
  fc2_in[(128 + ch) * NPTS + n] = sid;
}

// ---------------- final: current + mean_4( target_feat * x gathered ) ----------------
__global__ void final_kernel(const float* __restrict__ cur, const float* __restrict__ tgt,
                             const float* __restrict__ x, const int* __restrict__ inds_pp,
                             float* __restrict__ out) {
  int t = blockIdx.x * blockDim.x + threadIdx.x;
  if (t >= 160 * NPTS) return;
  int ch = t / NPTS, n = t % NPTS;
  float acc = 0.f;
  for (int j = 0; j < NPP; ++j) {
    int id = inds_pp[n * NPP + j];
    acc += tgt[ch * NPTS + id] * x[ch * NPTS + id];
  }
  out[ch * NPTS + n] = cur[ch * NPTS + n] + acc * (1.f / NPP);
}

extern "C" void kernel_launch(void* const* d_in, const int* in_sizes, int n_in,
                              void* d_out, int out_size, void* d_ws, size_t ws_size,
                              hipStream_t stream) {
  const float* img       = (const float*)d_in[0];   // (32,500)
  const float* cloud     = (const float*)d_in[1];   // (500,3)
  const float* img_tar   = (const float*)d_in[2];   // (32,500)
  const float* cloud_tar = (const float*)d_in[3];   // (500,3)
  const float* cur_feat  = (const float*)d_in[4];   // (160,500)
  const float* tgt_feat  = (const float*)d_in[5];   // (160,500)
  const float* W_conv1  = (const float*)d_in[6],  *b_conv1  = (const float*)d_in[7];
  const float* W_conv2  = (const float*)d_in[8],  *b_conv2  = (const float*)d_in[9];
  const float* W_pconv1 = (const float*)d_in[10], *b_pconv1 = (const float*)d_in[11];
  const float* W_pconv2 = (const float*)d_in[12], *b_pconv2 = (const float*)d_in[13];
  const float* W_fc1    = (const float*)d_in[14], *b_fc1    = (const float*)d_in[15];
  const float* W_fc2    = (const float*)d_in[16], *b_fc2    = (const float*)d_in[17];
  const float* W_fuse2  = (const float*)d_in[18], *b_fuse2  = (const float*)d_in[19];
  const float* W_pn1    = (const float*)d_in[20], *b_pn1    = (const float*)d_in[21];
  const float* W_pn2    = (const float*)d_in[22], *b_pn2    = (const float*)d_in[23];
  const float* W_pn3    = (const float*)d_in[24], *b_pn3    = (const float*)d_in[25];

  // -------- workspace carve (≈5.8 MB) --------
  char* base = (char*)d_ws; size_t off = 0;
  auto carve = [&](size_t bytes) -> void* {
    void* p = base + off; off = (off + bytes + 255) & ~(size_t)255; return p;
  };
  float* Wp      = (float*)carve(64 * 4 * 4);       // padded W_pconv1 (64x4)
  float* c_t     = (float*)carve(4 * NPTS * 4);     // (4,500), row3 = 0
  float* ct_t    = (float*)carve(4 * NPTS * 4);
  int* inds      = (int*)carve(NPTS * NP12 * 4);
  int* inds_self = (int*)carve(NPTS * NP12 * 4);
  int* inds_pp   = (int*)carve(NPTS * NPP * 4);
  float* w_ct    = (float*)carve(NPTS * NP12 * 4);
  float* w_cc    = (float*)carve(NPTS * NP12 * 4);
  float* A       = (float*)carve(128 * NPTS * 4);   // ifeat(img_tar)
  float* B       = (float*)carve(128 * NPTS * 4);   // ifeat(img)
  float* cf      = (float*)carve(128 * NPTS * 4);   // pfeat(cloud_tar)
  float* ctf     = (float*)carve(128 * NPTS * 4);   // pfeat(cloud)
  float* h64     = (float*)carve(64 * NPTS * 4);    // hidden (reused)
  float* fc1_in  = (float*)carve(256 * NPTS * 4);   // [img_diff; spd]
  float* fc2_in  = (float*)carve(256 * NPTS * 4);   // [cloud_diff; sid]
  float* fuse_in = (float*)carve(128 * NPTS * 4);   // [fuse_p; fuse_i]
  float* fuse_t  = (float*)carve(160 * NPTS * 4);
  float* pn1o    = (float*)carve(256 * NPTS * 4);
  float* pn2o    = (float*)carve(1024 * NPTS * 4);
  float* xbuf    = (float*)carve(160 * NPTS * 4);

  auto gemm = [&](const float* W, const float* X, const float* bias, float* Y,
                  int M, int K, int rowOff, int act) {
    int waves = (M / 16) * 8;                 // 8 strips of 64 cols
    int blocks = (waves + 7) / 8;             // 8 waves (256 threads) per block
#define GEMM_CASE(KK)                                                                     \
    case KK:                                                                              \
      if (act) gemm_wmma_kernel<KK, 1><<<blocks, 256, 0, stream>>>(W, X, bias, Y, M, rowOff); \
      else     gemm_wmma_kernel<KK, 0><<<blocks, 256, 0, stream>>>(W, X, bias, Y, M, rowOff); \
      break;
    switch (K) {
      GEMM_CASE(4)
      GEMM_CASE(32)
      GEMM_CASE(64)
      GEMM_CASE(128)
      GEMM_CASE(160)
      GEMM_CASE(256)
      GEMM_CASE(1024)
      default: break;
    }
#undef GEMM_CASE
  };

  prep_kernel<<<(64 * 4 + 4 * NPTS + 255) / 256, 256, 0, stream>>>(W_pconv1, cloud, cloud_tar,
                                                                   Wp, c_t, ct_t);
  knn_kernel<<<dim3(2, 3), 256, 0, stream>>>(cloud, cloud_tar, inds, inds_self, inds_pp);

  // pfeat(cloud_tar) -> cf ; pfeat(cloud) -> ctf   (K padded 3 -> 4)
  gemm(Wp, ct_t, b_pconv1, h64, 64, 4, 0, 1);
  gemm(W_pconv2, h64, b_pconv2, cf, 128, 64, 0, 0);
  gemm(Wp, c_t, b_pconv1, h64, 64, 4, 0, 1);
  gemm(W_pconv2, h64, b_pconv2, ctf, 128, 64, 0, 0);
  // ifeat(img_tar) -> A ; ifeat(img) -> B  (gather after conv == conv after gather)
  gemm(W_conv1, img_tar, b_conv1, h64, 64, 32, 0, 1);
  gemm(W_conv2, h64, b_conv2, A, 128, 64, 0, 0);
  gemm(W_conv1, img, b_conv1, h64, 64, 32, 0, 1);
  gemm(W_conv2, h64, b_conv2, B, 128, 64, 0, 0);

  softmax_w_kernel<<<2, 1024, 0, stream>>>(cloud, cloud_tar, inds, inds_self, w_ct, w_cc);

  gmax1_kernel<<<(128 * NPTS + 255) / 256, 256, 0, stream>>>(A, B, cf, ctf, inds, w_ct,
                                                             fc1_in, fc2_in);
  gmax2_kernel<<<(128 * NPTS + 255) / 256, 256, 0, stream>>>(inds_self, w_cc, fc1_in, fc2_in);

  gemm(W_fc1, fc1_in, b_fc1, fuse_in, 64, 256, 64, 0);  // fuse_i -> rows 64..127
  gemm(W_fc2, fc2_in, b_fc2, fuse_in, 64, 256, 0, 0);   // fuse_p -> rows 0..63
  gemm(W_fuse2, fuse_in, b_fuse2, fuse_t, 160, 128, 0, 0);
  gemm(W_pn1, fuse_t, b_pn1, pn1o, 256, 160, 0, 0);
  gemm(W_pn2, pn1o, b_pn2, pn2o, 1024, 256, 0, 1);      // leaky-relu
  gemm(W_pn3, pn2o, b_pn3, xbuf, 160, 1024, 0, 0);

  final_kernel<<<(160 * NPTS + 255) / 256, 256, 0, stream>>>(cur_feat, tgt_feat, xbuf,
                                                             inds_pp, (float*)d_out);
}